// Conv3x3_1417339207691
// MI455X (gfx1250) — compile-verified
//
#include <hip/hip_runtime.h>

// CDNA5 / gfx1250: wave32, V_WMMA_F32_16X16X4_F32 (D16x16 = A16x4 * B4x16 + C)
typedef __attribute__((ext_vector_type(2))) float v2f;
typedef __attribute__((ext_vector_type(8))) float v8f;

#define IMG_W   4608            // x is 4608 x 4608 f32
#define NPJ     1536            // patches per image row (stride-3)
#define OUT_PER_BLOCK 128       // 8 waves x 16 patches
// out = -dot(patch, w)/0.924458  -> fold scale into B (weights)
#define OUT_SCALE (-1.0f / 0.924458f)

// Each wave computes 16 consecutive outputs (patches along pj) as a GEMM:
//   A[m=0..15, k=0..11] = patch values (k = flattened 3x3, k>=9 zero-pad)
//   B[k, n]             = w_flat[k] * OUT_SCALE, replicated over all n
// using 3 chained v_wmma_f32_16x16x4_f32 (K = 4 each).
__global__ __launch_bounds__(256) void sc_conv3x3_s3_wmma(
    const float* __restrict__ x,
    const float* __restrict__ w,
    float* __restrict__ out)
{
    __shared__ float tile[OUT_PER_BLOCK];

    const int tid  = threadIdx.x;
    const int lane = tid & 31;
    const int wave = tid >> 5;
    const int l    = lane & 15;   // M index (patch within the wave's 16)
    const int half = lane >> 4;   // selects K pair: half0 -> k={4j,4j+1}, half1 -> k={4j+2,4j+3}

    const int m0 = blockIdx.x * OUT_PER_BLOCK + wave * 16;
    const int m  = m0 + l;              // global patch index (NPJ is a multiple of 16 -> no row split)
    const int pi = m / NPJ;
    const int pj = m - pi * NPJ;
    const float* base = x + (size_t)(3 * pi) * IMG_W + (size_t)(3 * pj);

    v8f acc = {0.f, 0.f, 0.f, 0.f, 0.f, 0.f, 0.f, 0.f};
    v2f a, b;

    // ---- WMMA 0 : k = 0..3  (w00 w01 | w02 w10) ----
    if (half == 0) {
        a[0] = base[0];                a[1] = base[1];
        b[0] = w[0] * OUT_SCALE;       b[1] = w[1] * OUT_SCALE;
    } else {
        a[0] = base[2];                a[1] = base[IMG_W + 0];
        b[0] = w[2] * OUT_SCALE;       b[1] = w[3] * OUT_SCALE;
    }
    acc = __builtin_amdgcn_wmma_f32_16x16x4_f32(false, a, false, b,
                                                (short)0, acc, false, false);

    // ---- WMMA 1 : k = 4..7  (w11 w12 | w20 w21) ----
    if (half == 0) {
        a[0] = base[IMG_W + 1];        a[1] = base[IMG_W + 2];
        b[0] = w[4] * OUT_SCALE;       b[1] = w[5] * OUT_SCALE;
    } else {
        a[0] = base[2 * IMG_W + 0];    a[1] = base[2 * IMG_W + 1];
        b[0] = w[6] * OUT_SCALE;       b[1] = w[7] * OUT_SCALE;
    }
    acc = __builtin_amdgcn_wmma_f32_16x16x4_f32(false, a, false, b,
                                                (short)0, acc, false, false);

    // ---- WMMA 2 : k = 8..11 (w22 0 | 0 0) ----
    if (half == 0) {
        a[0] = base[2 * IMG_W + 2];    a[1] = 0.f;
        b[0] = w[8] * OUT_SCALE;       b[1] = 0.f;
    } else {
        a[0] = 0.f;                    a[1] = 0.f;
        b[0] = 0.f;                    b[1] = 0.f;
    }
    acc = __builtin_amdgcn_wmma_f32_16x16x4_f32(false, a, false, b,
                                                (short)0, acc, false, false);

    // D layout: lane n (<16) VGPR v holds D[M=v, N=n]; lane n+16 holds D[M=8+v, N=n].
    // All N columns are identical (B replicated), so column 0 suffices:
    // lane 0 has outputs m0+0..7, lane 16 has m0+8..15. Stage via LDS for a
    // fully coalesced 128-float block store.
    if (l == 0) {
        float* dst = &tile[wave * 16 + half * 8];
        dst[0] = acc[0]; dst[1] = acc[1]; dst[2] = acc[2]; dst[3] = acc[3];
        dst[4] = acc[4]; dst[5] = acc[5]; dst[6] = acc[6]; dst[7] = acc[7];
    }
    __syncthreads();

    if (tid < OUT_PER_BLOCK) {
        out[(size_t)blockIdx.x * OUT_PER_BLOCK + tid] = tile[tid];
    }
}

extern "C" void kernel_launch(void* const* d_in, const int* in_sizes, int n_in,
                              void* d_out, int out_size, void* d_ws, size_t ws_size,
                              hipStream_t stream)
{
    const float* x = (const float*)d_in[0];   // 4608*4608 f32
    const float* w = (const float*)d_in[1];   // 3*3 f32
    float*     out = (float*)d_out;           // 1536*1536 f32

    const int blocks = out_size / OUT_PER_BLOCK;   // 2359296 / 128 = 18432 (exact)
    sc_conv3x3_s3_wmma<<<blocks, 256, 0, stream>>>(x, w, out);
}